// LightweightSwinEncoder3D_44315472560656
// MI455X (gfx1250) — compile-verified
//
#include <hip/hip_runtime.h>
#include <math.h>

// ---------------------------------------------------------------------------
// LightweightSwinEncoder3D for MI455X (gfx1250, wave32, WMMA).
// All GEMMs and the windowed attention use V_WMMA_F32_16X16X4_F32 so the
// math stays in fp32 (matches the fp32 reference) while running on the
// matrix pipe. Problem is bandwidth-bound (~15 GFLOP vs ~200MB traffic; the
// whole intermediate working set fits in the 192MB L2), so fp32 WMMA is the
// right precision/throughput tradeoff.
// ---------------------------------------------------------------------------

typedef __attribute__((ext_vector_type(2))) float v2f;
typedef __attribute__((ext_vector_type(8))) float v8f;

// ---------------------------------------------------------------------------
// Generic GEMM: C[M,N] = A[M,K] @ W[N,K]^T (+ bias[N]).
// One wave computes a 16x(16*NT) strip: the A fragment is loaded once per
// k-step and reused by NT WMMAs. NT=3 divides tilesN for every GEMM in this
// model (N in {48,96,144,192,288,384,576,768,1152}), so there is no
// remainder and control flow stays wave-uniform (EXEC all-ones for WMMA).
// M, N multiples of 16; K multiple of 4 (always true here).
// ---------------------------------------------------------------------------
template <int NT>
__global__ __launch_bounds__(128)
void gemm_wmma_f32(const float* __restrict__ A, const float* __restrict__ W,
                   const float* __restrict__ bias, float* __restrict__ C,
                   int M, int K, int N)
{
    int tilesN  = N >> 4;
    int groupsN = tilesN / NT;                   // exact: tilesN % NT == 0
    int waveId  = blockIdx.x * (blockDim.x >> 5) + (threadIdx.x >> 5);
    int lane    = threadIdx.x & 31;
    int groups  = (M >> 4) * groupsN;
    if (waveId >= groups) return;                // uniform per wave: EXEC stays full
    int tm = waveId / groupsN, tg = waveId - tm * groupsN;
    int m0 = tm << 4, n0 = tg * (NT << 4);
    int half = lane >> 4, r = lane & 15;

    // A 16x4 frag: lanes 0-15 hold K=k,k+1 ; lanes 16-31 hold K=k+2,k+3.
    // B 4x16 frag mirrors A with M<->N;  B[k][n] = W[n][k].
    const float* arow = A + (size_t)(m0 + r) * K + 2 * half;
    const float* wrow = W + (size_t)(n0 + r) * K + 2 * half;
    size_t wstride = (size_t)16 * K;

    v8f acc[NT];
#pragma unroll
    for (int nt = 0; nt < NT; ++nt) acc[nt] = (v8f){};

    for (int k = 0; k < K; k += 4) {
        v2f a;
        a.x = arow[k];
        a.y = arow[k + 1];
#pragma unroll
        for (int nt = 0; nt < NT; ++nt) {
            v2f b;
            const float* wr = wrow + (size_t)nt * wstride;
            b.x = wr[k];
            b.y = wr[k + 1];
            acc[nt] = __builtin_amdgcn_wmma_f32_16x16x4_f32(
                false, a, false, b, (short)0, acc[nt], false, false);
        }
        if (k + 64 < K)                           // global_prefetch_b8 one chunk ahead
            __builtin_prefetch(arow + k + 64, 0, 1);
    }
    // C/D layout: VGPR v -> row (v + 8*half), col = lane&15
#pragma unroll
    for (int nt = 0; nt < NT; ++nt) {
        int nc = n0 + (nt << 4) + r;
        float bn = bias ? bias[nc] : 0.0f;
        for (int v = 0; v < 8; ++v)
            C[(size_t)(m0 + v + 8 * half) * N + nc] = acc[nt][v] + bn;
    }
}

// ---------------------------------------------------------------------------
// Windowed attention: one wave per (batch, window, head).
// Q,K,V are 64x16 (head_dim is 16 at every stage), S is 64x64.
// S = (Q*0.25) @ K^T via 4x4 WMMA tiles (K=16), softmax rows in LDS,
// O = S @ V via 4 WMMA tiles (K=64). Gathers/scatters window tokens directly
// so no explicit window partition buffers are needed.
// ---------------------------------------------------------------------------
__global__ __launch_bounds__(32)
void window_attn(const float* __restrict__ qkv, float* __restrict__ out,
                 int Dm, int C, int nh)
{
    int nW  = Dm >> 2;
    int wpb = nW * nW * nW;
    int idx = blockIdx.x;
    int head = idx % nh;  idx /= nh;
    int win  = idx % wpb; int b = idx / wpb;
    int ww = win % nW, wh = (win / nW) % nW, wd = win / (nW * nW);

    int lane = threadIdx.x & 31;
    int half = lane >> 4, r = lane & 15;

    __shared__ float Qs[64][17];
    __shared__ float Ks[64][17];
    __shared__ float Vs[64][17];
    __shared__ float Ss[64][65];

    int C3 = 3 * C;
    int L  = Dm * Dm * Dm;
    size_t baseRow = (size_t)b * L;
    int hc = head * 16;

    for (int t = lane; t < 64; t += 32) {
        int id = t >> 4, ih = (t >> 2) & 3, iw = t & 3;
        int l = ((wd * 4 + id) * Dm + (wh * 4 + ih)) * Dm + (ww * 4 + iw);
        const float* row = qkv + (baseRow + l) * (size_t)C3;
        for (int c = 0; c < 16; ++c) {
            Qs[t][c] = row[hc + c] * 0.25f;        // hd^-0.5 = 1/sqrt(16)
            Ks[t][c] = row[C + hc + c];
            Vs[t][c] = row[2 * C + hc + c];
        }
    }
    __syncthreads();

    // S = Q @ K^T
    for (int mt = 0; mt < 4; ++mt) {
        for (int nt = 0; nt < 4; ++nt) {
            v8f acc = {};
            for (int k = 0; k < 16; k += 4) {
                v2f a, bf;
                a.x  = Qs[mt * 16 + r][k + 2 * half];
                a.y  = Qs[mt * 16 + r][k + 2 * half + 1];
                bf.x = Ks[nt * 16 + r][k + 2 * half];   // B[k][n] = K[n][k]
                bf.y = Ks[nt * 16 + r][k + 2 * half + 1];
                acc = __builtin_amdgcn_wmma_f32_16x16x4_f32(false, a, false, bf,
                                                            (short)0, acc, false, false);
            }
            for (int v = 0; v < 8; ++v)
                Ss[mt * 16 + v + 8 * half][nt * 16 + r] = acc[v];
        }
    }
    __syncthreads();

    // softmax over rows of S (2 rows per lane)
    for (int rowi = lane; rowi < 64; rowi += 32) {
        float mx = -1e30f;
        for (int j = 0; j < 64; ++j) mx = fmaxf(mx, Ss[rowi][j]);
        float sum = 0.0f;
        for (int j = 0; j < 64; ++j) {
            float e = __expf(Ss[rowi][j] - mx);
            Ss[rowi][j] = e;
            sum += e;
        }
        float inv = 1.0f / sum;
        for (int j = 0; j < 64; ++j) Ss[rowi][j] *= inv;
    }
    __syncthreads();

    // O = S @ V  (64x16), scatter back to token order
    for (int mt = 0; mt < 4; ++mt) {
        v8f acc = {};
        for (int k = 0; k < 64; k += 4) {
            v2f a, bf;
            a.x  = Ss[mt * 16 + r][k + 2 * half];
            a.y  = Ss[mt * 16 + r][k + 2 * half + 1];
            bf.x = Vs[k + 2 * half][r];             // B[k][n] = V[k][n]
            bf.y = Vs[k + 2 * half + 1][r];
            acc = __builtin_amdgcn_wmma_f32_16x16x4_f32(false, a, false, bf,
                                                        (short)0, acc, false, false);
        }
        for (int v = 0; v < 8; ++v) {
            int t = mt * 16 + v + 8 * half;
            int id = t >> 4, ih = (t >> 2) & 3, iw = t & 3;
            int l = ((wd * 4 + id) * Dm + (wh * 4 + ih)) * Dm + (ww * 4 + iw);
            out[(baseRow + l) * (size_t)C + hc + r] = acc[v];
        }
    }
}

// ---------------------------------------------------------------------------
// LayerNorm over last dim C: one block per row.
// ---------------------------------------------------------------------------
__global__ __launch_bounds__(128)
void layernorm_k(const float* __restrict__ in, float* __restrict__ out,
                 const float* __restrict__ g, const float* __restrict__ bb, int C)
{
    int row = blockIdx.x;
    const float* x = in + (size_t)row * C;
    float*       y = out + (size_t)row * C;
    __shared__ float red[128];
    float s = 0.f, s2 = 0.f;
    for (int c = threadIdx.x; c < C; c += blockDim.x) {
        float v = x[c]; s += v; s2 += v * v;
    }
    red[threadIdx.x] = s; __syncthreads();
    for (int st = 64; st > 0; st >>= 1) {
        if ((int)threadIdx.x < st) red[threadIdx.x] += red[threadIdx.x + st];
        __syncthreads();
    }
    float mean = red[0] / C; __syncthreads();
    red[threadIdx.x] = s2; __syncthreads();
    for (int st = 64; st > 0; st >>= 1) {
        if ((int)threadIdx.x < st) red[threadIdx.x] += red[threadIdx.x + st];
        __syncthreads();
    }
    float var = red[0] / C - mean * mean;
    float inv = rsqrtf(var + 1e-5f);
    for (int c = threadIdx.x; c < C; c += blockDim.x)
        y[c] = (x[c] - mean) * inv * g[c] + bb[c];
}

// ---------------------------------------------------------------------------
// Patch embed: 4x4x4 conv stride 4, 4->48 channels. One block per token.
// ---------------------------------------------------------------------------
__global__ __launch_bounds__(64)
void patch_embed_k(const float* __restrict__ x, const float* __restrict__ w,
                   const float* __restrict__ bias, float* __restrict__ y)
{
    int idx = blockIdx.x;                 // b*32768 + l
    int b = idx >> 15, l = idx & 32767;
    int d = l >> 10, h = (l >> 5) & 31, wq = l & 31;
    __shared__ float patch[256];
    const size_t plane = 128 * 128, vol = (size_t)128 * 128 * 128;
    const float* xb = x + (size_t)b * 4 * vol;
    for (int p = threadIdx.x; p < 256; p += 64) {
        int ci = p >> 6, dd = (p >> 4) & 3, hh = (p >> 2) & 3, wwq = p & 3;
        patch[p] = xb[(size_t)ci * vol + (size_t)(d * 4 + dd) * plane
                      + (size_t)(h * 4 + hh) * 128 + (wq * 4 + wwq)];
    }
    __syncthreads();
    if (threadIdx.x < 48) {
        int o = threadIdx.x;
        const float* wr = w + o * 256;    // pe_w [48][4][4][4][4] flattened
        float acc = bias[o];
        for (int p = 0; p < 256; ++p) acc = fmaf(patch[p], wr[p], acc);
        y[(size_t)idx * 48 + o] = acc;
    }
}

// ---------------------------------------------------------------------------
// Elementwise helpers
// ---------------------------------------------------------------------------
__global__ void add_k(float* __restrict__ t, const float* __restrict__ s, size_t n)
{
    size_t i = (size_t)blockIdx.x * blockDim.x + threadIdx.x;
    if (i < n) t[i] += s[i];
}

__global__ void gelu_k(float* __restrict__ d, size_t n)
{
    size_t i = (size_t)blockIdx.x * blockDim.x + threadIdx.x;
    if (i < n) {
        float v = d[i];
        d[i] = 0.5f * v * (1.0f + erff(v * 0.70710678118654752f));
    }
}

// t[B, Dm^3, C] -> out[B, D2^3, 8C] with concat index j = (w*2+h)*2+d
__global__ void merge_rearrange_k(const float* __restrict__ t, float* __restrict__ out,
                                  int Dm, int C, size_t total)
{
    size_t i = (size_t)blockIdx.x * blockDim.x + threadIdx.x;
    if (i >= total) return;
    int D2 = Dm >> 1;
    int c = (int)(i % C); size_t rest = i / C;
    int j = (int)(rest % 8); rest /= 8;
    int dd = j & 1, hh = (j >> 1) & 1, wwq = j >> 2;
    size_t l2 = rest % (size_t)(D2 * D2 * D2);
    int b = (int)(rest / (size_t)(D2 * D2 * D2));
    int w2 = (int)(l2 % D2), h2 = (int)((l2 / D2) % D2), d2 = (int)(l2 / (D2 * D2));
    int lin = ((d2 * 2 + dd) * Dm + (h2 * 2 + hh)) * Dm + (w2 * 2 + wwq);
    out[i] = t[((size_t)b * Dm * Dm * Dm + lin) * C + c];
}

// feature: [B, L, C] -> [B, C, L]
__global__ void feat_k(const float* __restrict__ t, float* __restrict__ out,
                       int L, int C, size_t total)
{
    size_t i = (size_t)blockIdx.x * blockDim.x + threadIdx.x;
    if (i >= total) return;
    int l = (int)(i % L);
    int c = (int)((i / L) % C);
    int b = (int)(i / ((size_t)L * C));
    out[i] = t[((size_t)b * L + l) * C + c];
}

// ---------------------------------------------------------------------------
// Host-side orchestration
// ---------------------------------------------------------------------------
struct BlockP {
    const float *fc1_b, *fc1_w, *fc2_b, *fc2_w, *n1b, *n1g, *n2b, *n2g,
                *proj_b, *proj_w, *qkv_b, *qkv_w;
};
struct MergeP { const float *nb, *ng, *red_w; };

extern "C" void kernel_launch(void* const* d_in, const int* in_sizes, int n_in,
                              void* d_out, int out_size, void* d_ws, size_t ws_size,
                              hipStream_t stream)
{
    (void)in_sizes; (void)n_in; (void)out_size; (void)ws_size;

    // ---- unpack inputs (jax pytree order: dict keys sorted, lists in order) ---
    int ii = 0;
    const float* x     = (const float*)d_in[ii++];
    const float* pe_b  = (const float*)d_in[ii++];
    const float* pe_nb = (const float*)d_in[ii++];
    const float* pe_ng = (const float*)d_in[ii++];
    const float* pe_w  = (const float*)d_in[ii++];
    BlockP blocks[4][2];
    MergeP merges[3];
    for (int s = 0; s < 4; ++s) {
        for (int bk = 0; bk < 2; ++bk) {
            BlockP& p = blocks[s][bk];
            p.fc1_b  = (const float*)d_in[ii++];
            p.fc1_w  = (const float*)d_in[ii++];
            p.fc2_b  = (const float*)d_in[ii++];
            p.fc2_w  = (const float*)d_in[ii++];
            p.n1b    = (const float*)d_in[ii++];
            p.n1g    = (const float*)d_in[ii++];
            p.n2b    = (const float*)d_in[ii++];
            p.n2g    = (const float*)d_in[ii++];
            p.proj_b = (const float*)d_in[ii++];
            p.proj_w = (const float*)d_in[ii++];
            p.qkv_b  = (const float*)d_in[ii++];
            p.qkv_w  = (const float*)d_in[ii++];
        }
        if (s < 3) {
            merges[s].nb    = (const float*)d_in[ii++];
            merges[s].ng    = (const float*)d_in[ii++];
            merges[s].red_w = (const float*)d_in[ii++];
        }
    }

    // ---- workspace layout (floats) ----
    float* ws   = (float*)d_ws;
    float* t    = ws;                      // [2, 32768, 48] max  = 3,145,728
    float* h    = t    + 3145728;          // scratch, same size
    float* qkvb = h    + 3145728;          // [2, 32768, 144]     = 9,437,184
    float* ao   = qkvb + 9437184;          // attn out / LN scratch = 3,145,728
    float* mlp  = ao   + 3145728;          // fc1 out max [2,32768,96] = 6,291,456

    auto gemm = [&](const float* A, const float* W, const float* bias, float* C,
                    int M, int K, int N) {
        int groups  = (M >> 4) * ((N >> 4) / 3);   // NT = 3 divides all tilesN
        int blocksN = (groups + 3) / 4;            // 4 waves / block
        gemm_wmma_f32<3><<<blocksN, 128, 0, stream>>>(A, W, bias, C, M, K, N);
    };
    auto ewb = [](size_t n) { return (unsigned)((n + 255) / 256); };

    const int NH[4] = {3, 6, 12, 24};

    // ---- patch embed + LN ----
    patch_embed_k<<<2 * 32768, 64, 0, stream>>>(x, pe_w, pe_b, h);
    layernorm_k<<<2 * 32768, 128, 0, stream>>>(h, t, pe_ng, pe_nb, 48);

    // ---- stages ----
    int Dm = 32, C = 48;
    float* outp = (float*)d_out;
    size_t featOff = 0;
    for (int s = 0; s < 4; ++s) {
        int L = Dm * Dm * Dm;
        int rows = 2 * L;
        int hdim = 2 * C;                  // MLP_RATIO = 2
        for (int bk = 0; bk < 2; ++bk) {
            const BlockP& p = blocks[s][bk];
            layernorm_k<<<rows, 128, 0, stream>>>(t, h, p.n1g, p.n1b, C);
            gemm(h, p.qkv_w, p.qkv_b, qkvb, rows, C, 3 * C);
            int nW = Dm >> 2;
            int attnBlocks = 2 * nW * nW * nW * NH[s];
            window_attn<<<attnBlocks, 32, 0, stream>>>(qkvb, ao, Dm, C, NH[s]);
            gemm(ao, p.proj_w, p.proj_b, h, rows, C, C);
            add_k<<<ewb((size_t)rows * C), 256, 0, stream>>>(t, h, (size_t)rows * C);
            layernorm_k<<<rows, 128, 0, stream>>>(t, h, p.n2g, p.n2b, C);
            gemm(h, p.fc1_w, p.fc1_b, mlp, rows, C, hdim);
            gelu_k<<<ewb((size_t)rows * hdim), 256, 0, stream>>>(mlp, (size_t)rows * hdim);
            gemm(mlp, p.fc2_w, p.fc2_b, h, rows, hdim, C);
            add_k<<<ewb((size_t)rows * C), 256, 0, stream>>>(t, h, (size_t)rows * C);
        }
        // feature i: [B, L, C] -> [B, C, D, H, W]
        size_t featN = (size_t)2 * L * C;
        feat_k<<<ewb(featN), 256, 0, stream>>>(t, outp + featOff, L, C, featN);
        featOff += featN;

        if (s < 3) {
            const MergeP& mp = merges[s];
            int D2 = Dm >> 1;
            int L2 = D2 * D2 * D2;
            int rows2 = 2 * L2;
            int C8 = 8 * C;
            size_t n8 = (size_t)rows2 * C8;
            merge_rearrange_k<<<ewb(n8), 256, 0, stream>>>(t, qkvb, Dm, C, n8);
            layernorm_k<<<rows2, 128, 0, stream>>>(qkvb, ao, mp.ng, mp.nb, C8);
            gemm(ao, mp.red_w, nullptr, t, rows2, C8, 2 * C);
            Dm = D2;
            C = 2 * C;
        }
    }
}